// ActivationHyperbolic_70480413327939
// MI455X (gfx1250) — compile-verified
//
#include <hip/hip_runtime.h>
#include <cstdint>

// ---------------------------------------------------------------------------
// ActivationHyperbolic (Poincare ball): y = proj(expmap0(relu(logmap0(x))))
// N x 64 fp32, N = 2M. Bandwidth-bound: ~1.07 GB traffic -> ~46us @ 23.3 TB/s.
// CDNA5 path: async global->LDS double-buffered streaming (ASYNCcnt +
// s_wait_asynccnt), wave32 16-lane butterflies, NT b128 stores, and hardware
// TRANS ops (v_sqrt/v_log/v_rcp/v_tanh) instead of libm expansions so VALU
// issue stays well under the HBM-roofline budget (~200 VALU per 2KB chunk).
// Math: s1,s2 > 0  =>  relu(s1*x) = s1*relu(x),  ||y|| = s1*s2*||relu(x)||,
// so one pass with two fused reductions (sum x^2, sum relu(x)^2) suffices.
// ---------------------------------------------------------------------------

typedef __attribute__((ext_vector_type(4))) float v4f;

#define MIN_NORM   1e-7f
#define BALL_EPS   4e-3f
#define ATANH_EPS  1e-7f

#define WAVES_PER_BLOCK 8
#define BLOCK_THREADS   (WAVES_PER_BLOCK * 32)
#define CHUNK_FLOATS    256   // 4 rows x 64 floats = 1KB per chunk
#define CHUNK_BYTES     1024
#define NBUF            2

__device__ __forceinline__ float frcp(float a)  { return __builtin_amdgcn_rcpf(a); }
__device__ __forceinline__ float fsqrt(float a) { return __builtin_amdgcn_sqrtf(a); }

// log2 via hardware v_log_f32 (guarded; falls back to libm)
__device__ __forceinline__ float flog2(float a) {
#if __has_builtin(__builtin_amdgcn_logf)
    return __builtin_amdgcn_logf(a);
#else
    return __log2f(a);
#endif
}

// tanh(z)/z for z >= MIN_NORM
__device__ __forceinline__ float tanh_over_z(float z) {
#if __has_builtin(__builtin_amdgcn_tanhf)
    return __builtin_amdgcn_tanhf(z) * frcp(z);          // v_tanh_f32 (gfx1250)
#elif __has_builtin(__builtin_amdgcn_exp2f)
    // small z: 1 - z^2/3 + 2z^4/15 (avoids cancellation in exp form)
    float z2    = z * z;
    float small = fmaf(z2, fmaf(z2, 0.13333334f, -0.33333334f), 1.0f);
    float e     = __builtin_amdgcn_exp2f(z * 2.8853901f); // exp(2z)
    float big   = (1.0f - 2.0f * frcp(e + 1.0f)) * frcp(z);
    return (z < 0.25f) ? small : big;
#else
    return tanhf(z) / z;
#endif
}

// Per-row scalar chain: given ss = sum x^2, pp = sum relu(x)^2, sqrt(c),
// return the single positive scale applied to relu(x).
__device__ __forceinline__ float final_scale(float ss, float pp, float sc) {
    float xn   = fmaxf(fsqrt(ss), MIN_NORM);              // clamped ||x||
    float scxn = sc * xn;
    float arg  = fminf(scxn, 1.0f - ATANH_EPS);
    // atanh(a) = 0.5*ln2 * log2((1+a)/(1-a))
    float at   = 0.34657359f * flog2((1.0f + arg) * frcp(1.0f - arg));
    float s1   = at * frcp(scxn);                         // logmap0 scale (>0)
    float rn   = fsqrt(pp);                               // ||relu(x)||
    float un   = fmaxf(s1 * rn, MIN_NORM);                // clamped ||u||
    float z    = sc * un;
    float s2   = tanh_over_z(z);                          // expmap0 scale (>0)
    float yn   = fmaxf(s1 * s2 * rn, MIN_NORM);           // clamped ||y||
    float mx   = (1.0f - BALL_EPS) * frcp(sc);            // ball radius
    float fs   = s1 * s2;
    if (yn > mx) fs *= mx * frcp(yn);                     // proj into ball
    return fs;
}

// One 512B half-chunk: 16 lanes per row, 4 floats per lane, wave32.
__device__ __forceinline__ v4f body(v4f v, float sc) {
    float px = fmaxf(v.x, 0.0f), py = fmaxf(v.y, 0.0f);
    float pz = fmaxf(v.z, 0.0f), pw = fmaxf(v.w, 0.0f);
    float ss = v.x * v.x + v.y * v.y + v.z * v.z + v.w * v.w;
    float pp = px * px + py * py + pz * pz + pw * pw;
    // Butterfly within each 16-lane half of the wave32 (row group).
#pragma unroll
    for (int m = 1; m <= 8; m <<= 1) {
        ss += __shfl_xor(ss, m, 32);
        pp += __shfl_xor(pp, m, 32);
    }
    float fs = final_scale(ss, pp, sc);
    v4f r = {px * fs, py * fs, pz * fs, pw * fs};
    return r;
}

__global__ __launch_bounds__(BLOCK_THREADS)
void hyper_act_async(const float* __restrict__ x, const float* __restrict__ c,
                     float* __restrict__ out, unsigned long long nchunks)
{
    __shared__ __align__(16) unsigned char smem[WAVES_PER_BLOCK * NBUF * CHUNK_BYTES];

    const unsigned lane = threadIdx.x & 31u;
    const unsigned wv   = threadIdx.x >> 5;
    const unsigned long long w0 = (unsigned long long)blockIdx.x * WAVES_PER_BLOCK + wv;
    const unsigned long long ws = (unsigned long long)gridDim.x * WAVES_PER_BLOCK;

    unsigned char* wbase = smem + wv * (NBUF * CHUNK_BYTES);
    // Low 32 bits of the generic LDS pointer == workgroup-relative LDS byte
    // address, which is what the async-copy VDST operand wants.
    const uint32_t ldsLane0 = (uint32_t)(uintptr_t)wbase + lane * 16u;

    const float sc = sqrtf(c[0]);

    // Prologue: prefetch first chunk (2 x b128 per lane -> ASYNCcnt += 2).
    if (w0 < nchunks) {
        const float* g = x + w0 * (unsigned long long)CHUNK_FLOATS + lane * 4u;
        asm volatile(
            "global_load_async_to_lds_b128 %0, %1, off th:TH_LOAD_NT\n\t"
            "global_load_async_to_lds_b128 %0, %1, off offset:512 th:TH_LOAD_NT"
            :: "v"(ldsLane0), "v"((unsigned long long)(uintptr_t)g)
            : "memory");
    }

    int cur = 0;
    for (unsigned long long ch = w0; ch < nchunks; ch += ws) {
        unsigned long long nx = ch + ws;
        if (nx < nchunks) {
            // Prefetch next chunk into the other buffer, then wait until the
            // current chunk's two (oldest) async copies have landed.
            const float* g = x + nx * (unsigned long long)CHUNK_FLOATS + lane * 4u;
            asm volatile(
                "global_load_async_to_lds_b128 %0, %1, off th:TH_LOAD_NT\n\t"
                "global_load_async_to_lds_b128 %0, %1, off offset:512 th:TH_LOAD_NT"
                :: "v"(ldsLane0 + (uint32_t)((cur ^ 1) * CHUNK_BYTES)),
                   "v"((unsigned long long)(uintptr_t)g)
                : "memory");
            asm volatile("s_wait_asynccnt 2" ::: "memory");
        } else {
            asm volatile("s_wait_asynccnt 0" ::: "memory");
        }

        const unsigned char* b = wbase + cur * CHUNK_BYTES + lane * 16u;
        v4f v0 = *(const v4f*)(b);          // rows 4ch, 4ch+1
        v4f v1 = *(const v4f*)(b + 512);    // rows 4ch+2, 4ch+3

        v4f r0 = body(v0, sc);
        v4f r1 = body(v1, sc);

        float* o = out + ch * (unsigned long long)CHUNK_FLOATS + lane * 4u;
        __builtin_nontemporal_store(r0, (v4f*)o);          // write-once stream
        __builtin_nontemporal_store(r1, (v4f*)(o + 128));
        cur ^= 1;
    }
}

// Scalar fallback for rows % 4 (not hit for the reference 2M x 64 shape).
__global__ void hyper_act_tail(const float* __restrict__ x, const float* __restrict__ c,
                               float* __restrict__ out,
                               unsigned long long row0, unsigned long long rows)
{
    unsigned long long r = row0 + (unsigned long long)blockIdx.x * blockDim.x + threadIdx.x;
    if (r >= rows) return;
    const float* xr = x + r * 64ull;
    float sc = sqrtf(c[0]);
    float ss = 0.0f, pp = 0.0f;
    for (int i = 0; i < 64; ++i) {
        float a = xr[i];
        float p = fmaxf(a, 0.0f);
        ss = fmaf(a, a, ss);
        pp = fmaf(p, p, pp);
    }
    float fs = final_scale(ss, pp, sc);
    float* orow = out + r * 64ull;
    for (int i = 0; i < 64; ++i) orow[i] = fmaxf(xr[i], 0.0f) * fs;
}

extern "C" void kernel_launch(void* const* d_in, const int* in_sizes, int n_in,
                              void* d_out, int out_size, void* d_ws, size_t ws_size,
                              hipStream_t stream)
{
    const float* x = (const float*)d_in[0];
    const float* c = (const float*)d_in[1];
    float* out = (float*)d_out;

    unsigned long long total = (unsigned long long)in_sizes[0]; // N * 64
    unsigned long long rows  = total / 64ull;
    unsigned long long nchunks = rows / 4ull;                   // 4 rows / chunk
    unsigned long long tail0   = nchunks * 4ull;

    if (nchunks > 0) {
        unsigned long long blk = (nchunks + WAVES_PER_BLOCK - 1) / WAVES_PER_BLOCK;
        if (blk > 4096ull) blk = 4096ull;   // 32K persistent waves, grid-stride
        hipLaunchKernelGGL(hyper_act_async, dim3((unsigned)blk), dim3(BLOCK_THREADS),
                           0, stream, x, c, out, nchunks);
    }
    if (tail0 < rows) {
        unsigned tailRows = (unsigned)(rows - tail0);
        hipLaunchKernelGGL(hyper_act_tail, dim3((tailRows + 63) / 64), dim3(64),
                           0, stream, x, c, out, tail0, rows);
    }
}